// Net_63565515981322
// MI455X (gfx1250) — compile-verified
//
#include <hip/hip_runtime.h>
#include <math.h>

// ---------------------------------------------------------------------------
// JacobiPool GCN forward for MI455X (gfx1250, wave32, WMMA).
// GEMMs (h @ convW, lin1, lin2) use V_WMMA_F32_16X16X4_F32 with double-buffered
// LDS staging via the CDNA5 async-to-LDS engine (global_load_async_to_lds_b128,
// ASYNCcnt) software-pipelined against the WMMA loop.
// Graph propagation = f32 atomic scatter over edges (L2 atomics).
// Workspace requirement: ~96 MB.
// ---------------------------------------------------------------------------

#define B_GR   128
#define N0     512
#define E_TOT  (128 * 4096)     // 524288 edges (static; pooled edges get w=0)
#define H_DIM  128
#define AJ     1.0f
#define BJ     1.0f
#define EPSBN  1e-5f
#define KC     64               // GEMM K-chunk staged in LDS
#define CHUNK_F (64 * KC + KC * 64)   // floats per LDS buffer (A strip + B panel)

typedef __attribute__((ext_vector_type(2))) float v2f;
typedef __attribute__((ext_vector_type(8))) float v8f;

static inline int cdiv(int a, int b) { return (a + b - 1) / b; }

// ---------------------------------------------------------------- utilities
__global__ void k_zero_f(float* __restrict__ p, int n) {
    int i = blockIdx.x * blockDim.x + threadIdx.x;
    if (i < n) p[i] = 0.f;
}

__global__ void k_fill_i32(int* __restrict__ p, int v, int n) {
    int i = blockIdx.x * blockDim.x + threadIdx.x;
    if (i < n) p[i] = v;
}

__global__ void k_scale_f(float* __restrict__ p, float s, int n) {
    int i = blockIdx.x * blockDim.x + threadIdx.x;
    if (i < n) p[i] *= s;
}

// edge state init: private copies of src/dst + unit weights
__global__ void k_init_edges(const int* __restrict__ s_in, const int* __restrict__ d_in,
                             int* __restrict__ s, int* __restrict__ d,
                             float* __restrict__ w, int E) {
    int e = blockIdx.x * blockDim.x + threadIdx.x;
    if (e >= E) return;
    s[e] = s_in[e];
    d[e] = d_in[e];
    w[e] = 1.f;
}

// deg[dst] += w  (deg pre-zeroed)
__global__ void k_deg_edge(const int* __restrict__ dst, const float* __restrict__ w,
                           float* __restrict__ deg, int E) {
    int e = blockIdx.x * blockDim.x + threadIdx.x;
    if (e >= E) return;
    float we = w[e];
    if (we != 0.f) atomicAdd(&deg[dst[e]], we);
}

// deg -> dinv = rsqrt(deg + 1)   (self loop)
__global__ void k_dinv(float* __restrict__ deg, int N) {
    int i = blockIdx.x * blockDim.x + threadIdx.x;
    if (i < N) deg[i] = rsqrtf(deg[i] + 1.f);
}

__global__ void k_enorm(const int* __restrict__ src, const int* __restrict__ dst,
                        const float* __restrict__ w, const float* __restrict__ dinv,
                        float* __restrict__ enorm, int E) {
    int e = blockIdx.x * blockDim.x + threadIdx.x;
    if (e >= E) return;
    enorm[e] = dinv[src[e]] * dinv[dst[e]] * w[e];
}

// -------------------------------------------------------------- WMMA GEMM
// C[M x N] = A[M x K] * B[K x N], row-major f32. Requires M%64==0, N%64==0
// (N is 64 or 128 everywhere in this net), K%KC==0.
//
// Block = 128 threads = 4 waves; block computes a 64x64 macro-tile.
// K is processed in KC-deep chunks. Each chunk's A strip (64xKC) and B panel
// (KCx64) are staged into LDS with global_load_async_to_lds_b128 (ASYNCcnt),
// double-buffered and software-pipelined: while chunk c is consumed by WMMAs,
// chunk c+1 is already in flight to the other LDS buffer.
//
// Per k-step all 5 LDS fragments (1 B + 4 A) are loaded before the 4
// back-to-back v_wmma_f32_16x16x4_f32 ops (disjoint accumulators -> no
// WMMA-to-WMMA hazards, one dscnt wait covers the fragment group).
//
// Register layouts per CDNA5 ISA 7.12.2 (32-bit operands):
//   A 16x4 : lanes 0-15 m=0..15 {v0:K+0, v1:K+1}; lanes 16-31 {v0:K+2, v1:K+3}
//   B 4x16 : lanes hold n=lane&15; v0/v1 hold rows (K+2*kq)/(K+2*kq+1)
//   C 16x16: lane n=lane&15; vgpr v holds row v + 8*(lane>>4)
__global__ void k_gemm_wmma(const float* __restrict__ A, const float* __restrict__ B,
                            float* __restrict__ C, int M, int K, int N) {
    extern __shared__ float lds[];        // 2 x (As[64][KC] ++ Bs[KC][64])

    const int tid  = threadIdx.x;
    const int m0   = blockIdx.y * 64;
    const int nblk = blockIdx.x * 64;
    const int lane = tid & 31;
    const int wv   = tid >> 5;            // wave id 0..3 -> 16-col N tile
    const int mrow = lane & 15;
    const int kq   = lane >> 4;           // 0: lanes 0-15, 1: lanes 16-31

    // issue async stage of one K-chunk into LDS buffer `buf` (no wait here)
    auto stage = [&](int k0, float* bufA) {
        float* bufB = bufA + 64 * KC;
        for (int i = tid; i < 2048; i += 128) {
            unsigned lds_off;
            unsigned long long ga;
            if (i < 1024) {               // A: row r (64), 16 float4 per row
                int r = i >> 4, c = (i & 15) << 2;
                lds_off = (unsigned)(size_t)(void*)&bufA[r * KC + c];
                ga = (unsigned long long)(const void*)
                     (A + (size_t)(m0 + r) * K + k0 + c);
            } else {                      // B: row r (KC), 16 float4 per row
                int j = i - 1024;
                int r = j >> 4, c = (j & 15) << 2;
                lds_off = (unsigned)(size_t)(void*)&bufB[r * 64 + c];
                ga = (unsigned long long)(const void*)
                     (B + (size_t)(k0 + r) * N + nblk + c);
            }
            asm volatile("global_load_async_to_lds_b128 %0, %1, off"
                         :: "v"(lds_off), "v"(ga) : "memory");
        }
    };

    v8f acc0 = {0.f,0.f,0.f,0.f,0.f,0.f,0.f,0.f};
    v8f acc1 = acc0, acc2 = acc0, acc3 = acc0;

    const int nch = K / KC;
    stage(0, lds);                        // prime the pipeline

    for (int c = 0; c < nch; ++c) {
        asm volatile("s_wait_asynccnt 0x0" ::: "memory");
        __syncthreads();                  // chunk c resident for all waves;
                                          // also: chunk c-1 compute finished,
                                          // safe to restage its buffer
        if (c + 1 < nch)                  // prefetch next chunk (other buffer)
            stage((c + 1) * KC, lds + ((c + 1) & 1) * CHUNK_F);

        const float* As = lds + (c & 1) * CHUNK_F;
        const float* Bp = As + 64 * KC + wv * 16 + (lane & 15);
        const float* Ap = As + mrow * KC;

        for (int k = 0; k < KC; k += 4) {
            const int kk = k + 2 * kq;
            // load all fragments first, then issue 4 WMMAs back-to-back
            v2f b, a0, a1, a2, a3;
            b.x  = Bp[kk * 64];              b.y  = Bp[(kk + 1) * 64];
            a0.x = Ap[kk];                   a0.y = Ap[kk + 1];
            a1.x = Ap[16 * KC + kk];         a1.y = Ap[16 * KC + kk + 1];
            a2.x = Ap[32 * KC + kk];         a2.y = Ap[32 * KC + kk + 1];
            a3.x = Ap[48 * KC + kk];         a3.y = Ap[48 * KC + kk + 1];
            acc0 = __builtin_amdgcn_wmma_f32_16x16x4_f32(false, a0, false, b, (short)0, acc0, false, false);
            acc1 = __builtin_amdgcn_wmma_f32_16x16x4_f32(false, a1, false, b, (short)0, acc1, false, false);
            acc2 = __builtin_amdgcn_wmma_f32_16x16x4_f32(false, a2, false, b, (short)0, acc2, false, false);
            acc3 = __builtin_amdgcn_wmma_f32_16x16x4_f32(false, a3, false, b, (short)0, acc3, false, false);
        }
    }

    // ---- store 4 C tiles ----
    const int ccol = nblk + wv * 16 + (lane & 15);
    float* Cp0 = C + (size_t)(m0 +  0 + 8 * kq) * N + ccol;
    float* Cp1 = C + (size_t)(m0 + 16 + 8 * kq) * N + ccol;
    float* Cp2 = C + (size_t)(m0 + 32 + 8 * kq) * N + ccol;
    float* Cp3 = C + (size_t)(m0 + 48 + 8 * kq) * N + ccol;
#pragma unroll
    for (int v = 0; v < 8; ++v) {
        Cp0[v * N] = acc0[v];
        Cp1[v * N] = acc1[v];
        Cp2[v * N] = acc2[v];
        Cp3[v * N] = acc3[v];
    }
}

// ---------------------------------------------------------- GCN propagation
// out[n,f] = tmp[n,f] * dinv[n]^2 + bias[f]          (self-loop + bias)
__global__ void k_prop_node(const float* __restrict__ tmp, const float* __restrict__ dinv,
                            const float* __restrict__ bias, float* __restrict__ out,
                            int total) {
    int i = blockIdx.x * blockDim.x + threadIdx.x;
    if (i >= total) return;
    int n = i >> 7, f = i & (H_DIM - 1);
    float dv = dinv[n];
    out[i] = tmp[i] * dv * dv + bias[f];
}

// out[dst[e], f] += enorm[e] * tmp[src[e], f]   — one block (128 thr) per edge
__global__ void k_prop_edge(const float* __restrict__ tmp, const int* __restrict__ src,
                            const int* __restrict__ dst, const float* __restrict__ enorm,
                            float* __restrict__ out) {
    int e = blockIdx.x;
    float en = enorm[e];
    if (en == 0.f) return;
    int s = src[e], d = dst[e], f = threadIdx.x;
    atomicAdd(&out[d * H_DIM + f], en * tmp[s * H_DIM + f]);
}

// ------------------------------------------------------------------ BatchNorm
__global__ void k_bn_stats(const float* __restrict__ h, float* __restrict__ mean,
                           float* __restrict__ var, int R, int Cc) {
    __shared__ float ss[256], sq[256];
    const int f = blockIdx.x, tid = threadIdx.x;
    float s = 0.f, q = 0.f;
    for (int i = tid; i < R; i += 256) { float v = h[i * Cc + f]; s += v; q += v * v; }
    ss[tid] = s; sq[tid] = q; __syncthreads();
    for (int off = 128; off > 0; off >>= 1) {
        if (tid < off) { ss[tid] += ss[tid + off]; sq[tid] += sq[tid + off]; }
        __syncthreads();
    }
    if (tid == 0) { float m = ss[0] / R; mean[f] = m; var[f] = sq[0] / R - m * m; }
}

__global__ void k_bn_relu(float* __restrict__ h, const float* __restrict__ mean,
                          const float* __restrict__ var, const float* __restrict__ g,
                          const float* __restrict__ b, int total, int Cc) {
    int i = blockIdx.x * blockDim.x + threadIdx.x;
    if (i >= total) return;
    int f = i % Cc;
    float v = (h[i] - mean[f]) * rsqrtf(var[f] + EPSBN) * g[f] + b[f];
    h[i] = fmaxf(v, 0.f);
}

__global__ void k_bias(float* __restrict__ h, const float* __restrict__ b,
                       int total, int Cc) {
    int i = blockIdx.x * blockDim.x + threadIdx.x;
    if (i < total) h[i] += b[i % Cc];
}

// ----------------------------------------------------------- attention score
// s0[n] = dot(h[n,:], att)  — one wave per node
__global__ void k_dot_att(const float* __restrict__ h, const float* __restrict__ att,
                          float* __restrict__ s0, int N) {
    const int lane = threadIdx.x & 31;
    const int node = blockIdx.x * 4 + (threadIdx.x >> 5);
    if (node >= N) return;
    const float* hp = h + node * H_DIM;
    float s = hp[lane]      * att[lane]      + hp[lane + 32] * att[lane + 32]
            + hp[lane + 64] * att[lane + 64] + hp[lane + 96] * att[lane + 96];
    for (int m = 16; m >= 1; m >>= 1) s += __shfl_xor(s, m, 32);
    if (lane == 0) s0[node] = s;
}

// ------------------------------------------------- Jacobi polynomial (pvec)
__global__ void k_pv_init(const float* __restrict__ v, const float* __restrict__ dinv,
                          float* __restrict__ pv, int N) {
    int i = blockIdx.x * blockDim.x + threadIdx.x;
    if (i >= N) return;
    float dv = dinv[i];
    pv[i] = v[i] * dv * dv;
}

__global__ void k_pv_edge(const float* __restrict__ v, const int* __restrict__ src,
                          const int* __restrict__ dst, const float* __restrict__ enorm,
                          float* __restrict__ pv, int E) {
    int e = blockIdx.x * blockDim.x + threadIdx.x;
    if (e >= E) return;
    float en = enorm[e];
    if (en != 0.f) atomicAdd(&pv[dst[e]], en * v[src[e]]);
}

__global__ void k_axpby(const float* __restrict__ x, const float* __restrict__ y,
                        float* __restrict__ out, float ca, float cb, int N) {
    int i = blockIdx.x * blockDim.x + threadIdx.x;
    if (i < N) out[i] = ca * x[i] + cb * y[i];
}

// out = a1*pv + a2*pcur - a3*pprev
__global__ void k_jacobi(const float* __restrict__ pv, const float* __restrict__ pcur,
                         const float* __restrict__ pprev, float* __restrict__ out,
                         float a1, float a2, float a3, int N) {
    int i = blockIdx.x * blockDim.x + threadIdx.x;
    if (i < N) out[i] = a1 * pv[i] + a2 * pcur[i] - a3 * pprev[i];
}

__global__ void k_score(const float* __restrict__ p0, const float* __restrict__ p1,
                        const float* __restrict__ p2, const float* __restrict__ p3,
                        const float* __restrict__ th, float* __restrict__ sc, int N) {
    int i = blockIdx.x * blockDim.x + threadIdx.x;
    if (i >= N) return;
    sc[i] = th[0] * p0[i] + th[1] * p1[i] + th[2] * p2[i] + th[3] * p3[i];
}

// ------------------------------------------------------- per-graph top-k
// One block of n threads per graph; shared-memory bitonic sort (descending).
__global__ void k_topk(const float* __restrict__ sc, int n, int* __restrict__ sel) {
    __shared__ float ssc[512];
    __shared__ int   sid[512];
    const int b = blockIdx.x, tid = threadIdx.x;
    ssc[tid] = sc[b * n + tid];
    sid[tid] = tid;
    __syncthreads();
    for (int ksz = 2; ksz <= n; ksz <<= 1) {
        for (int j = ksz >> 1; j > 0; j >>= 1) {
            int ixj = tid ^ j;
            if (ixj > tid) {
                bool desc = ((tid & ksz) == 0);
                float a = ssc[tid], c = ssc[ixj];
                bool sw = desc ? (a < c) : (a > c);
                if (sw) {
                    ssc[tid] = c; ssc[ixj] = a;
                    int t = sid[tid]; sid[tid] = sid[ixj]; sid[ixj] = t;
                }
            }
            __syncthreads();
        }
    }
    int k = n >> 1;
    if (tid < k) sel[b * k + tid] = b * n + sid[tid];
}

__global__ void k_set_newid(const int* __restrict__ sel, int* __restrict__ nid, int BK) {
    int i = blockIdx.x * blockDim.x + threadIdx.x;
    if (i < BK) nid[sel[i]] = i;
}

// hout[i,:] = h[sel[i],:] * tanh(score[sel[i]])   — block(128) per kept node
__global__ void k_gather(const float* __restrict__ h, const float* __restrict__ sc,
                         const int* __restrict__ sel, float* __restrict__ hout) {
    int i = blockIdx.x;
    int s = sel[i];
    float t = tanhf(sc[s]);
    hout[i * H_DIM + threadIdx.x] = h[s * H_DIM + threadIdx.x] * t;
}

__global__ void k_remap(int* __restrict__ src, int* __restrict__ dst,
                        float* __restrict__ w, const int* __restrict__ nid, int E) {
    int e = blockIdx.x * blockDim.x + threadIdx.x;
    if (e >= E) return;
    int ns = nid[src[e]], nd = nid[dst[e]];
    bool valid = (ns >= 0) && (nd >= 0) && (w[e] > 0.f);
    w[e]   = valid ? 1.f : 0.f;
    src[e] = valid ? ns : 0;
    dst[e] = valid ? nd : 0;
}

// reads[b, f] += max_j h[b,j,f];  reads[b, 128+f] += mean_j h[b,j,f]
__global__ void k_readout(const float* __restrict__ h, float* __restrict__ reads, int k) {
    int b = blockIdx.x, f = threadIdx.x;
    float mx = -3.402823466e+38f, sm = 0.f;
    const float* hp = h + (size_t)b * k * H_DIM + f;
    for (int j = 0; j < k; ++j) { float v = hp[j * H_DIM]; mx = fmaxf(mx, v); sm += v; }
    reads[b * 256 + f]       += mx;
    reads[b * 256 + 128 + f] += sm * (1.f / k);
}

// ----------------------------------------------------------------- MLP tail
__global__ void k_lin3(const float* __restrict__ g2, const float* __restrict__ W,
                       const float* __restrict__ b, float* __restrict__ logits) {
    int r = blockIdx.x, t = threadIdx.x;
    if (t >= 10) return;
    float s = b[t];
    const float* gp = g2 + r * 64;
    for (int j = 0; j < 64; ++j) s += gp[j] * W[j * 10 + t];
    logits[r * 10 + t] = s;
}

__global__ void k_logsoftmax(const float* __restrict__ logits, float* __restrict__ out) {
    int r = blockIdx.x, t = threadIdx.x;
    float z = (t < 10) ? logits[r * 10 + t] : -3.402823466e+38f;
    float mx = z;
    for (int m = 16; m >= 1; m >>= 1) mx = fmaxf(mx, __shfl_xor(mx, m, 32));
    float e = (t < 10) ? __expf(z - mx) : 0.f;
    float s = e;
    for (int m = 16; m >= 1; m >>= 1) s += __shfl_xor(s, m, 32);
    if (t < 10) out[r * 10 + t] = z - mx - __logf(s);
}

// ---------------------------------------------------------------------------
extern "C" void kernel_launch(void* const* d_in, const int* in_sizes, int n_in,
                              void* d_out, int out_size, void* d_ws, size_t ws_size,
                              hipStream_t stream) {
    (void)in_sizes; (void)n_in; (void)out_size; (void)ws_size;
    const float* x     = (const float*)d_in[0];
    const float* convW = (const float*)d_in[1];
    const float* convb = (const float*)d_in[2];
    const float* bnW   = (const float*)d_in[3];
    const float* bnB   = (const float*)d_in[4];
    const float* bn7W  = (const float*)d_in[5];
    const float* bn7B  = (const float*)d_in[6];
    const float* attW  = (const float*)d_in[7];
    const float* theta = (const float*)d_in[8];
    const float* lin1W = (const float*)d_in[9];
    const float* lin1b = (const float*)d_in[10];
    const float* lin2W = (const float*)d_in[11];
    const float* lin2b = (const float*)d_in[12];
    const float* lin3W = (const float*)d_in[13];
    const float* lin3b = (const float*)d_in[14];
    const int*   src0  = (const int*)d_in[15];
    const int*   dst0  = (const int*)d_in[16];
    float* out = (float*)d_out;

    // ------------------------- workspace carve (bump allocator, 256B align)
    char* p = (char*)d_ws;
    auto alloc = [&](size_t bytes) -> char* {
        char* r = p;
        p += (bytes + 255) & ~(size_t)255;
        return r;
    };
    const int NTOT = B_GR * N0;                     // 65536
    float* hA    = (float*)alloc((size_t)NTOT * H_DIM * 4);       // 32 MB
    float* hT    = (float*)alloc((size_t)NTOT * H_DIM * 4);       // 32 MB (GEMM out)
    float* hB    = (float*)alloc((size_t)(NTOT / 2) * H_DIM * 4); // 16 MB
    float* dinv  = (float*)alloc(NTOT * 4);         // deg -> dinv
    float* enorm = (float*)alloc(E_TOT * 4);
    float* wgt   = (float*)alloc(E_TOT * 4);
    int*   esrc  = (int*)  alloc(E_TOT * 4);
    int*   edst  = (int*)  alloc(E_TOT * 4);
    float* s0    = (float*)alloc(NTOT * 4);
    float* p1    = (float*)alloc(NTOT * 4);
    float* p2    = (float*)alloc(NTOT * 4);
    float* p3    = (float*)alloc(NTOT * 4);
    float* pv    = (float*)alloc(NTOT * 4);
    float* score = (float*)alloc(NTOT * 4);
    int*   sel   = (int*)  alloc((NTOT / 2) * 4);
    int*   nid   = (int*)  alloc(NTOT * 4);
    float* reads = (float*)alloc(B_GR * 256 * 4);
    float* mean  = (float*)alloc(256 * 4);
    float* var   = (float*)alloc(256 * 4);
    float* g1    = (float*)alloc(128 * 128 * 4);
    float* g2    = (float*)alloc(128 * 64 * 4);
    float* lgts  = (float*)alloc(128 * 16 * 4);

    const int EB = cdiv(E_TOT, 256);                // edge-parallel blocks
    const size_t GEMM_LDS = (size_t)2 * CHUNK_F * 4;   // 64 KB (double buffer)

    k_zero_f    <<<cdiv(B_GR * 256, 256), 256, 0, stream>>>(reads, B_GR * 256);
    k_init_edges<<<EB, 256, 0, stream>>>(src0, dst0, esrc, edst, wgt, E_TOT);

    // Jacobi recurrence coefficients (A_J = B_J = 1)
    auto jcoef = [](int kk, float& a1, float& a2, float& a3) {
        float A = AJ, B = BJ;
        float c0 = 2.f * kk * (kk + A + B) * (2.f * kk + A + B - 2.f);
        float c1 = (2.f * kk + A + B - 1.f) * (2.f * kk + A + B) * (2.f * kk + A + B - 2.f);
        float c2 = (2.f * kk + A + B - 1.f) * (A * A - B * B);
        float c3 = 2.f * (kk + A - 1.f) * (kk + B - 1.f) * (2.f * kk + A + B);
        a1 = c1 / c0; a2 = c2 / c0; a3 = c3 / c0;
    };
    float j2a, j2b, j2c, j3a, j3b, j3c;
    jcoef(2, j2a, j2b, j2c);
    jcoef(3, j3a, j3b, j3c);

    int n = N0;
    const float* hin = x;       // current node features (read-only this layer)
    float* cur = hA;            // propagation output (becomes h)
    float* nxt = hB;            // gather destination

    for (int layer = 0; layer < 5; ++layer) {
        const int N  = B_GR * n;
        const int NH = N * H_DIM;

        // --- symmetric normalization ---
        k_zero_f <<<cdiv(N, 256), 256, 0, stream>>>(dinv, N);
        k_deg_edge<<<EB, 256, 0, stream>>>(edst, wgt, dinv, E_TOT);
        k_dinv   <<<cdiv(N, 256), 256, 0, stream>>>(dinv, N);
        k_enorm  <<<EB, 256, 0, stream>>>(esrc, edst, wgt, dinv, enorm, E_TOT);

        // --- GCNConv: tmp = h @ W, then propagate + bias ---
        dim3 gg(H_DIM / 64, N / 64);
        k_gemm_wmma<<<gg, 128, GEMM_LDS, stream>>>(hin, convW + layer * H_DIM * H_DIM,
                                                   hT, N, H_DIM, H_DIM);
        k_prop_node<<<cdiv(NH, 256), 256, 0, stream>>>(hT, dinv, convb + layer * H_DIM, cur, NH);
        k_prop_edge<<<E_TOT, H_DIM, 0, stream>>>(hT, esrc, edst, enorm, cur);

        // --- BN + ReLU ---
        k_bn_stats<<<H_DIM, 256, 0, stream>>>(cur, mean, var, N, H_DIM);
        k_bn_relu <<<cdiv(NH, 256), 256, 0, stream>>>(cur, mean, var,
                                                      bnW + layer * H_DIM, bnB + layer * H_DIM,
                                                      NH, H_DIM);

        // --- JacobiPool scores ---
        k_dot_att<<<cdiv(N, 4), 128, 0, stream>>>(cur, attW + layer * H_DIM, s0, N);
        // p1 = 0.5*((A-B)*s0 + (A+B+2)*pvec(s0))
        k_pv_init<<<cdiv(N, 256), 256, 0, stream>>>(s0, dinv, pv, N);
        k_pv_edge<<<EB, 256, 0, stream>>>(s0, esrc, edst, enorm, pv, E_TOT);
        k_axpby  <<<cdiv(N, 256), 256, 0, stream>>>(s0, pv, p1,
                                                    0.5f * (AJ - BJ), 0.5f * (AJ + BJ + 2.f), N);
        // p2
        k_pv_init<<<cdiv(N, 256), 256, 0, stream>>>(p1, dinv, pv, N);
        k_pv_edge<<<EB, 256, 0, stream>>>(p1, esrc, edst, enorm, pv, E_TOT);
        k_jacobi <<<cdiv(N, 256), 256, 0, stream>>>(pv, p1, s0, p2, j2a, j2b, j2c, N);
        // p3
        k_pv_init<<<cdiv(N, 256), 256, 0, stream>>>(p2, dinv, pv, N);
        k_pv_edge<<<EB, 256, 0, stream>>>(p2, esrc, edst, enorm, pv, E_TOT);
        k_jacobi <<<cdiv(N, 256), 256, 0, stream>>>(pv, p2, p1, p3, j3a, j3b, j3c, N);
        // score = sum_j theta[layer,j] * p_j
        k_score  <<<cdiv(N, 256), 256, 0, stream>>>(s0, p1, p2, p3, theta + layer * 4, score, N);

        // --- top-k selection (k = n/2), gather, edge remap ---
        const int k  = n / 2;
        const int BK = B_GR * k;
        k_topk     <<<B_GR, n, 0, stream>>>(score, n, sel);
        k_fill_i32 <<<cdiv(N, 256), 256, 0, stream>>>(nid, -1, N);
        k_set_newid<<<cdiv(BK, 256), 256, 0, stream>>>(sel, nid, BK);
        k_gather   <<<BK, H_DIM, 0, stream>>>(cur, score, sel, nxt);
        k_remap    <<<EB, 256, 0, stream>>>(esrc, edst, wgt, nid, E_TOT);

        // --- readout (max || mean), accumulated across layers ---
        k_readout<<<B_GR, H_DIM, 0, stream>>>(nxt, reads, k);

        // rotate buffers: new h lives in `nxt`; next layer overwrites it in-place
        // (safe: GEMM fully consumes h before prop writes), gathers to the other.
        hin = nxt;
        float* t = cur; cur = nxt; nxt = t;
        n = k;
    }

    // ------------------------------ MLP head ------------------------------
    k_scale_f<<<cdiv(B_GR * 256, 256), 256, 0, stream>>>(reads, 0.2f, B_GR * 256);

    // g1 = relu(bn5(reads @ lin1W + lin1b))        [128x256]x[256x128]
    k_gemm_wmma<<<dim3(2, 2), 128, GEMM_LDS, stream>>>(reads, lin1W, g1, 128, 256, 128);
    k_bias    <<<cdiv(128 * 128, 256), 256, 0, stream>>>(g1, lin1b, 128 * 128, 128);
    k_bn_stats<<<128, 256, 0, stream>>>(g1, mean, var, 128, 128);
    k_bn_relu <<<cdiv(128 * 128, 256), 256, 0, stream>>>(g1, mean, var,
                                                         bnW + 5 * H_DIM, bnB + 5 * H_DIM,
                                                         128 * 128, 128);
    // g2 = relu(bn7(g1 @ lin2W + lin2b))           [128x128]x[128x64]
    k_gemm_wmma<<<dim3(1, 2), 128, GEMM_LDS, stream>>>(g1, lin2W, g2, 128, 128, 64);
    k_bias    <<<cdiv(128 * 64, 256), 256, 0, stream>>>(g2, lin2b, 128 * 64, 64);
    k_bn_stats<<<64, 256, 0, stream>>>(g2, mean, var, 128, 64);
    k_bn_relu <<<cdiv(128 * 64, 256), 256, 0, stream>>>(g2, mean, var, bn7W, bn7B,
                                                        128 * 64, 64);
    // logits + log_softmax
    k_lin3      <<<128, 64, 0, stream>>>(g2, lin3W, lin3b, lgts);
    k_logsoftmax<<<128, 32, 0, stream>>>(lgts, out);
}